// DrugDiseaseModel_86036784873727
// MI455X (gfx1250) — compile-verified
//
#include <hip/hip_runtime.h>

// Problem constants (match the reference)
#define NNODES 50000
#define NREL   20
#define EMB    64
#define HID    128

typedef __attribute__((ext_vector_type(2))) float v2f;
typedef __attribute__((ext_vector_type(8))) float v8f;

// ---------------------------------------------------------------------------
// Zero a buffer, float4 per thread
__global__ void zero_f4(float* __restrict__ p, size_t n4) {
  size_t i = (size_t)blockIdx.x * blockDim.x + threadIdx.x;
  if (i < n4) ((float4*)p)[i] = float4{0.f, 0.f, 0.f, 0.f};
}

// Per-(dst,rel) edge counts (same for both layers)
__global__ void count_edges(const int* __restrict__ dst, const int* __restrict__ et,
                            float* __restrict__ cnt, int n_edges) {
  int e = blockIdx.x * blockDim.x + threadIdx.x;
  if (e < n_edges)
    atomicAdd(cnt + (size_t)dst[e] * NREL + et[e], 1.0f);
}

// Scatter x[src] into agg[(dst*NREL+etype)*D + :] with float atomics.
// D=64 -> 16 lanes/edge (2 edges per wave32); D=128 -> 32 lanes/edge.
template <int D>
__global__ void scatter_edges(const int* __restrict__ src, const int* __restrict__ dst,
                              const int* __restrict__ et, const float* __restrict__ x,
                              float* __restrict__ agg, int n_edges) {
  constexpr int LPE = D / 4;        // lanes per edge (float4 each)
  constexpr int EPW = 32 / LPE;     // edges per wave
  size_t gtid = (size_t)blockIdx.x * blockDim.x + threadIdx.x;
  size_t wave = gtid >> 5;
  int lane = (int)(gtid & 31);
  int sub  = lane / LPE;            // edge slot within wave
  int l    = lane % LPE;            // lane within edge
  size_t e = wave * EPW + sub;
  if (e >= (size_t)n_edges) return;
  int s = src[e], d = dst[e], r = et[e];
  float4 v = *(const float4*)(x + (size_t)s * D + l * 4);
  float* p = agg + ((size_t)d * NREL + r) * D + l * 4;
  atomicAdd(p + 0, v.x);
  atomicAdd(p + 1, v.y);
  atomicAdd(p + 2, v.z);
  atomicAdd(p + 3, v.w);
}

// agg[seg, :] /= max(cnt[seg], 1)
template <int D>
__global__ void scale_mean(float* __restrict__ agg, const float* __restrict__ cnt,
                           size_t nseg) {
  constexpr int Q = D / 4;
  size_t i = (size_t)blockIdx.x * blockDim.x + threadIdx.x;
  if (i >= nseg * Q) return;
  size_t s   = i / Q;
  size_t off = (i % Q) * 4;
  float inv = 1.0f / fmaxf(cnt[s], 1.0f);
  float4* p = (float4*)(agg + s * (size_t)D + off);
  float4 v = *p;
  v.x *= inv; v.y *= inv; v.z *= inv; v.w *= inv;
  *p = v;
}

// ---------------------------------------------------------------------------
// out[n,o] = sum_k mean[n, k] * W[k, o]   (k over NREL*DIN, mean row contiguous)
//          + sum_i x[n, i]   * root[i, o]
//          + bias[o], optional ReLU.
// One block = 16 rows (nodes); 8 wave32s each own a 16-wide column tile
// (HID = 8*16 = 128). fp32 WMMA 16x16x4, two interleaved accumulators.
//
// A frag (16x4 f32): lane holds row M = lane&15; lanes 0-15 carry K=k,k+1 in
// v[0],v[1]; lanes 16-31 carry K=k+2,k+3. (ISA 7.12.2)
// B frag (4x16 f32): row-striped like C/D: VGPR0 = rows {k, k+2} across lane
// halves, VGPR1 = rows {k+1, k+3}.
// C/D (16x16 f32): VGPR v, lanes 0-15 -> M=v, lanes 16-31 -> M=v+8.
template <int DIN>
__global__ void __launch_bounds__(256)
rgcn_gemm(const float* __restrict__ mean, const float* __restrict__ xin,
          const float* __restrict__ W, const float* __restrict__ root,
          const float* __restrict__ bias, float* __restrict__ out, int relu) {
  constexpr int K1 = NREL * DIN;
  const int wave = threadIdx.x >> 5;
  const int lane = threadIdx.x & 31;
  const int hl = lane >> 4;      // lane half: selects K pair / M offset
  const int lm = lane & 15;      // row (for A) / column (for B,C,D)
  const int row0 = blockIdx.x * 16;
  const int col0 = wave * 16;

  const float* arow = mean + (size_t)(row0 + lm) * K1;
  const float* xrow = xin + (size_t)(row0 + lm) * DIN;

  v8f acc0 = {};
  v8f acc1 = {};

  // Relation part: mean[n, :] @ Wstack[K1, HID]
  for (int k = 0; k < K1; k += 8) {
    v2f a0 = *(const v2f*)(arow + k + 2 * hl);
    const float* bp0 = W + (size_t)(k + 2 * hl) * HID + col0 + lm;
    v2f b0; b0.x = bp0[0]; b0.y = bp0[HID];
    acc0 = __builtin_amdgcn_wmma_f32_16x16x4_f32(false, a0, false, b0,
                                                 (short)0, acc0, false, false);
    v2f a1 = *(const v2f*)(arow + k + 4 + 2 * hl);
    const float* bp1 = W + (size_t)(k + 4 + 2 * hl) * HID + col0 + lm;
    v2f b1; b1.x = bp1[0]; b1.y = bp1[HID];
    acc1 = __builtin_amdgcn_wmma_f32_16x16x4_f32(false, a1, false, b1,
                                                 (short)0, acc1, false, false);
  }
  // Root part: x[n, :] @ root[DIN, HID]
  for (int k = 0; k < DIN; k += 8) {
    v2f a0 = *(const v2f*)(xrow + k + 2 * hl);
    const float* bp0 = root + (size_t)(k + 2 * hl) * HID + col0 + lm;
    v2f b0; b0.x = bp0[0]; b0.y = bp0[HID];
    acc0 = __builtin_amdgcn_wmma_f32_16x16x4_f32(false, a0, false, b0,
                                                 (short)0, acc0, false, false);
    v2f a1 = *(const v2f*)(xrow + k + 4 + 2 * hl);
    const float* bp1 = root + (size_t)(k + 4 + 2 * hl) * HID + col0 + lm;
    v2f b1; b1.x = bp1[0]; b1.y = bp1[HID];
    acc1 = __builtin_amdgcn_wmma_f32_16x16x4_f32(false, a1, false, b1,
                                                 (short)0, acc1, false, false);
  }

  const float bv = bias[col0 + lm];
#pragma unroll
  for (int v = 0; v < 8; ++v) {
    float val = acc0[v] + acc1[v] + bv;
    if (relu) val = fmaxf(val, 0.f);
    out[(size_t)(row0 + v + 8 * hl) * HID + col0 + lm] = val;
  }
}

// ---------------------------------------------------------------------------
// DistMult: scores[b] = sum_o h[head[b],o] * rel_emb[rel[b],o] * h[tail[b],o]
__global__ void distmult(const float* __restrict__ h, const int* __restrict__ head,
                         const int* __restrict__ tail, const int* __restrict__ rel,
                         const float* __restrict__ rel_emb, float* __restrict__ out,
                         int batch) {
  size_t gtid = (size_t)blockIdx.x * blockDim.x + threadIdx.x;
  int b = (int)(gtid >> 5);
  int lane = (int)(gtid & 31);
  if (b >= batch) return;
  const float* hp = h + (size_t)head[b] * HID;
  const float* tp = h + (size_t)tail[b] * HID;
  const float* rp = rel_emb + (size_t)rel[b] * HID;
  float s = 0.f;
#pragma unroll
  for (int i = lane; i < HID; i += 32) s += hp[i] * rp[i] * tp[i];
#pragma unroll
  for (int off = 16; off > 0; off >>= 1) s += __shfl_xor(s, off, 32);
  if (lane == 0) out[b] = s;
}

// ---------------------------------------------------------------------------
static inline unsigned cdiv(size_t a, size_t b) { return (unsigned)((a + b - 1) / b); }

extern "C" void kernel_launch(void* const* d_in, const int* in_sizes, int n_in,
                              void* d_out, int out_size, void* d_ws, size_t ws_size,
                              hipStream_t stream) {
  const int*   edge_index = (const int*)d_in[0];   // [2, E]
  const int*   edge_type  = (const int*)d_in[1];   // [E]
  const int*   head_idx   = (const int*)d_in[2];   // [B]
  const int*   tail_idx   = (const int*)d_in[3];
  const int*   rel_types  = (const int*)d_in[4];
  const float* node_emb   = (const float*)d_in[5]; // [N, 64]
  const float* W1         = (const float*)d_in[6]; // [20, 64, 128]
  const float* root1      = (const float*)d_in[7]; // [64, 128]
  const float* b1         = (const float*)d_in[8];
  const float* W2         = (const float*)d_in[9]; // [20, 128, 128]
  const float* root2      = (const float*)d_in[10];
  const float* b2         = (const float*)d_in[11];
  const float* rel_emb    = (const float*)d_in[12];

  const int n_edges = in_sizes[1];
  const int batch   = in_sizes[2];
  const int* src = edge_index;
  const int* dst = edge_index + n_edges;

  // Workspace layout (floats)
  const size_t nseg  = (size_t)NNODES * NREL;          // 1e6 segments
  const size_t aggN  = nseg * HID;                     // 128M floats (layer-2 size)
  float* agg = (float*)d_ws;                           // reused by both layers
  float* cnt = agg + aggN;                             // [nseg]
  float* h1  = cnt + nseg;                             // [N, HID]
  float* h2  = h1 + (size_t)NNODES * HID;              // [N, HID]

  const int T = 256;

  // Edge counts (shared by both layers)
  zero_f4<<<cdiv(nseg / 4, T), T, 0, stream>>>(cnt, nseg / 4);
  count_edges<<<cdiv(n_edges, T), T, 0, stream>>>(dst, edge_type, cnt, n_edges);

  // ---- Layer 1 (DIN = 64) ----
  {
    const size_t n1 = nseg * EMB;
    zero_f4<<<cdiv(n1 / 4, T), T, 0, stream>>>(agg, n1 / 4);
    size_t lanes = ((size_t)(n_edges + 1) / 2) * 32;   // 2 edges per wave
    scatter_edges<EMB><<<cdiv(lanes, T), T, 0, stream>>>(src, dst, edge_type,
                                                         node_emb, agg, n_edges);
    scale_mean<EMB><<<cdiv(nseg * (EMB / 4), T), T, 0, stream>>>(agg, cnt, nseg);
    rgcn_gemm<EMB><<<NNODES / 16, 256, 0, stream>>>(agg, node_emb, W1, root1, b1,
                                                    h1, /*relu=*/1);
  }

  // ---- Layer 2 (DIN = 128) ----
  {
    const size_t n2 = nseg * HID;
    zero_f4<<<cdiv(n2 / 4, T), T, 0, stream>>>(agg, n2 / 4);
    size_t lanes = (size_t)n_edges * 32;               // 1 edge per wave
    scatter_edges<HID><<<cdiv(lanes, T), T, 0, stream>>>(src, dst, edge_type,
                                                         h1, agg, n_edges);
    scale_mean<HID><<<cdiv(nseg * (HID / 4), T), T, 0, stream>>>(agg, cnt, nseg);
    rgcn_gemm<HID><<<NNODES / 16, 256, 0, stream>>>(agg, h1, W2, root2, b2,
                                                    h2, /*relu=*/0);
  }

  // ---- DistMult decoder ----
  distmult<<<cdiv((size_t)batch * 32, T), T, 0, stream>>>(h2, head_idx, tail_idx,
                                                          rel_types, rel_emb,
                                                          (float*)d_out, batch);
}